// ParticleModel_33998961115178
// MI455X (gfx1250) — compile-verified
//
#include <hip/hip_runtime.h>

typedef __attribute__((ext_vector_type(2))) float v2f;
typedef __attribute__((ext_vector_type(8))) float v8f;

#define NUM_PARTICLES 8192
#define NUM_OBJECTS   2
#define GRID          30
#define NPTS          (GRID * GRID * GRID)   /* 27000 */
#define MTILES        ((NPTS + 15) / 16)     /* 1688  */
#define NTILES        (NUM_PARTICLES / 16)   /* 512 particle tiles */

// c = (1 / (2*sigma^2)) * log2(e) = 200 * 1.4426950408889634
#define CSCALE 288.53900817779268f
// norm = 1 / (sqrt(2*pi) * sigma * P)
#define NORM   (1.0f / (2.5066282746310002f * 0.05f * 8192.0f))

// ---------------------------------------------------------------------------
// Prep: warp particle z, prescale, and store the WMMA B-operand in *lane
// order* so the main loop needs zero per-iteration selects.
// Entry [obj][tile][lane] (float4):
//   lane  0-15 (half 0): { 2c*px, 2c*py, -c*p_sq, 0 }   (B rows K0,K1)
//   lane 16-31 (half 1): { 2c*zt, -c,    -c*p_sq, 0 }   (B rows K2,K3)
// where the particle is n = tile*16 + (lane&15).
// ---------------------------------------------------------------------------
__global__ __launch_bounds__(256) void kde_prep(const float* __restrict__ particles,
                                                float4* __restrict__ bout) {
    int i = blockIdx.x * blockDim.x + threadIdx.x;
    if (i >= NUM_OBJECTS * NTILES * 32) return;
    int obj  = i / (NTILES * 32);
    int rem  = i % (NTILES * 32);
    int t    = rem / 32;
    int lane = rem % 32;
    int n    = t * 16 + (lane & 15);

    const float* p = particles + (size_t)n * (3 * NUM_OBJECTS) + 3 * obj;
    float px = p[0], py = p[1], pz = p[2];
    float zt  = copysignf(sqrtf(fabsf(pz)), pz);
    float psq = px * px + py * py + zt * zt;

    float4 v;
    if (lane < 16) {
        v = make_float4(2.0f * CSCALE * px, 2.0f * CSCALE * py, -CSCALE * psq, 0.0f);
    } else {
        v = make_float4(2.0f * CSCALE * zt, -CSCALE, -CSCALE * psq, 0.0f);
    }
    bout[i] = v;
}

// ---------------------------------------------------------------------------
// Main KDE: one wave32 per (object, 16-grid-row tile). Loops 512 particle
// tiles; each iteration: b128 load -> V_WMMA_F32_16X16X4_F32 -> 8x v_exp_f32.
// ---------------------------------------------------------------------------
__global__ __launch_bounds__(256) void kde_main(const float* __restrict__ vol,
                                                const float4* __restrict__ bdata,
                                                float* __restrict__ out) {
    const int lane = threadIdx.x & 31;
    const int wid  = blockIdx.x * 8 + (threadIdx.x >> 5);
    if (wid >= NUM_OBJECTS * MTILES) return;   // wave-uniform (never taken at this grid)

    const int obj = wid / MTILES;
    const int mt  = wid % MTILES;
    const int m0  = mt * 16;

    // --- A matrix (16x4 f32): row M = [x, y, z, v_sq].
    // ISA layout: lanes 0-15 M=0..15 {V0=K0, V1=K1}; lanes 16-31 {V0=K2, V1=K3}.
    const int  mrow   = m0 + (lane & 15);
    const int  mclamp = (mrow < NPTS) ? mrow : (NPTS - 1);
    const float x = vol[(size_t)mclamp * 3 + 0];
    const float y = vol[(size_t)mclamp * 3 + 1];
    const float z = vol[(size_t)mclamp * 3 + 2];
    const float vsq = x * x + y * y + z * z;
    const int  h = lane >> 4;
    v2f A;
    A.x = h ? z   : x;   // K = 2h
    A.y = h ? vsq : y;   // K = 2h+1

    const float4* bp = bdata + (size_t)obj * NTILES * 32;

    v8f acc = {0.f, 0.f, 0.f, 0.f, 0.f, 0.f, 0.f, 0.f};

    for (int t = 0; t < NTILES; ++t) {
        // Pre-selected, lane-ordered B operand: no cndmask in the loop.
        const float4 b = bp[t * 32 + lane];
        v2f B;
        B.x = b.x;
        B.y = b.y;
        const float addend = b.z;   // -c * p_sq for column n = lane&15

        v8f czero = {0.f, 0.f, 0.f, 0.f, 0.f, 0.f, 0.f, 0.f};
        // D[m,n] = 2c*dot(v_m, p_n) - c*v_sq[m]
        v8f d = __builtin_amdgcn_wmma_f32_16x16x4_f32(
            /*neg_a=*/false, A, /*neg_b=*/false, B,
            /*c_mod=*/(short)0, czero, /*reuse_a=*/false, /*reuse_b=*/false);

#pragma unroll
        for (int r = 0; r < 8; ++r) {
            float arg = d[r] + addend;            // = -c * d2  (pre-clamp)
            arg = (arg < 0.0f) ? arg : 0.0f;      // d2 = max(d2, 0)
            acc[r] += __builtin_amdgcn_exp2f(arg);
        }
    }

    // --- Row sums: reduce each 16-lane half (xor masks 1,2,4,8 stay in-half).
#pragma unroll
    for (int r = 0; r < 8; ++r) {
        float v = acc[r];
        v += __shfl_xor(v, 1, 32);
        v += __shfl_xor(v, 2, 32);
        v += __shfl_xor(v, 4, 32);
        v += __shfl_xor(v, 8, 32);
        acc[r] = v;
    }

    // D layout: VGPR r -> row M = r (lanes 0-15) or M = r+8 (lanes 16-31).
    if ((lane & 15) == 0) {
        const int mb = m0 + h * 8;
#pragma unroll
        for (int r = 0; r < 8; ++r) {
            const int m = mb + r;
            if (m < NPTS) out[(size_t)obj * NPTS + m] = acc[r] * NORM;
        }
    }
}

// ---------------------------------------------------------------------------
// Fallback (tiny ws): identical math, particle transform done inline.
// ---------------------------------------------------------------------------
__global__ __launch_bounds__(256) void kde_main_inline(const float* __restrict__ vol,
                                                       const float* __restrict__ particles,
                                                       float* __restrict__ out) {
    const int lane = threadIdx.x & 31;
    const int wid  = blockIdx.x * 8 + (threadIdx.x >> 5);
    if (wid >= NUM_OBJECTS * MTILES) return;

    const int obj = wid / MTILES;
    const int mt  = wid % MTILES;
    const int m0  = mt * 16;

    const int  mrow   = m0 + (lane & 15);
    const int  mclamp = (mrow < NPTS) ? mrow : (NPTS - 1);
    const float x = vol[(size_t)mclamp * 3 + 0];
    const float y = vol[(size_t)mclamp * 3 + 1];
    const float z = vol[(size_t)mclamp * 3 + 2];
    const float vsq = x * x + y * y + z * z;
    const int  h = lane >> 4;
    v2f A;
    A.x = h ? z   : x;
    A.y = h ? vsq : y;

    v8f acc = {0.f, 0.f, 0.f, 0.f, 0.f, 0.f, 0.f, 0.f};

    for (int t = 0; t < NTILES; ++t) {
        const int n = t * 16 + (lane & 15);
        const float* p = particles + (size_t)n * (3 * NUM_OBJECTS) + 3 * obj;
        float px = p[0], py = p[1], pz = p[2];
        float zt  = copysignf(sqrtf(fabsf(pz)), pz);
        float psq = px * px + py * py + zt * zt;

        v2f B;
        B.x = h ? (2.0f * CSCALE * zt) : (2.0f * CSCALE * px);
        B.y = h ? -CSCALE              : (2.0f * CSCALE * py);
        const float addend = -CSCALE * psq;

        v8f czero = {0.f, 0.f, 0.f, 0.f, 0.f, 0.f, 0.f, 0.f};
        v8f d = __builtin_amdgcn_wmma_f32_16x16x4_f32(
            false, A, false, B, (short)0, czero, false, false);

#pragma unroll
        for (int r = 0; r < 8; ++r) {
            float arg = d[r] + addend;
            arg = (arg < 0.0f) ? arg : 0.0f;
            acc[r] += __builtin_amdgcn_exp2f(arg);
        }
    }

#pragma unroll
    for (int r = 0; r < 8; ++r) {
        float v = acc[r];
        v += __shfl_xor(v, 1, 32);
        v += __shfl_xor(v, 2, 32);
        v += __shfl_xor(v, 4, 32);
        v += __shfl_xor(v, 8, 32);
        acc[r] = v;
    }

    if ((lane & 15) == 0) {
        const int mb = m0 + h * 8;
#pragma unroll
        for (int r = 0; r < 8; ++r) {
            const int m = mb + r;
            if (m < NPTS) out[(size_t)obj * NPTS + m] = acc[r] * NORM;
        }
    }
}

extern "C" void kernel_launch(void* const* d_in, const int* in_sizes, int n_in,
                              void* d_out, int out_size, void* d_ws, size_t ws_size,
                              hipStream_t stream) {
    const float* particles = (const float*)d_in[0];   // (8192, 6) f32
    const float* volume    = (const float*)d_in[1];   // (30,30,30,3) f32
    float* out             = (float*)d_out;           // (2, 30,30,30) f32

    const int nwaves = NUM_OBJECTS * MTILES;          // 3376
    const int blocks = (nwaves + 7) / 8;              // 8 waves / 256-thread block

    const size_t need = (size_t)NUM_OBJECTS * NTILES * 32 * sizeof(float4);  // 512 KB
    if (ws_size >= need) {
        float4* bdata = (float4*)d_ws;
        const int prepN = NUM_OBJECTS * NTILES * 32;
        kde_prep<<<(prepN + 255) / 256, 256, 0, stream>>>(particles, bdata);
        kde_main<<<blocks, 256, 0, stream>>>(volume, bdata, out);
    } else {
        kde_main_inline<<<blocks, 256, 0, stream>>>(volume, particles, out);
    }
}